// AugmentedUKF_81389630259824
// MI455X (gfx1250) — compile-verified
//
#include <hip/hip_runtime.h>
#include <cstdint>

// ---------------- UKF constants (from reference) ----------------
#define DTs    0.05f
#define CDAMP  0.25f
#define AP_    0.1f
#define AV_    0.1f
#define JIT    1e-6f
#define SQRT_C 1.6f          // sqrt(2.56)
#define WM0    (-0.5625f)    // LAM/CUT
#define WMI    0.1953125f    // 1/(2*CUT)
#define WC0    1.7975f       // WM0 + (1 - 0.64 + 2)
#define WCI    0.1953125f

#define TLEN   256
#define TT     8             // time steps per LDS tile
#define NTILE  (TLEN / TT)
#define BLK    256

// LDS y-tile: 256 rows x (TT*2 floats = 64B data) padded to 80B (16B aligned
// for async b128 stores; consumer ds_load_b64 sees at most 2-way conflicts)
#define YROW   20            // floats per row

// async loads issued per thread per tile (4 x b128 = one 64B y row slice each)
#define AEPT   4

typedef float v4f __attribute__((ext_vector_type(4)));   // native vector for NT store

__global__ __launch_bounds__(BLK) void ukf_kernel(
    const float* __restrict__ y, const float* __restrict__ u,
    const float* __restrict__ z0, const float* __restrict__ P0,
    float* __restrict__ out)
{
  __shared__ float ldsY[2][BLK * YROW];

  const int tid = threadIdx.x;
  const int b0  = blockIdx.x * BLK;
  const int b   = b0 + tid;

  // Issue async copy of y-tile k (steps [k*TT, k*TT+TT)) into buffer k&1.
  // Each thread moves 4 float4s; a wave covers whole 64B rows (coalesced).
  auto issue_tile = [&](int k) {
    const int t0 = k * TT;
    float* dstBase = &ldsY[k & 1][0];
    #pragma unroll
    for (int i = 0; i < AEPT; ++i) {
      int f = i * BLK + tid;                 // float4 index in tile
      int r = f >> 2, c = f & 3;             // row (batch), 16B chunk in row
      uint32_t lo = (uint32_t)(uintptr_t)dstBase + (uint32_t)(r * (YROW * 4) + c * 16);
      const uint8_t* ga = (const uint8_t*)y +
          ((size_t)(b0 + r) * TLEN + (size_t)t0) * 8u + (size_t)(c * 16);
      asm volatile("global_load_async_to_lds_b128 %0, %1, off"
                   :: "v"(lo), "v"(ga) : "memory");
    }
  };

  // ---------------- load initial state ----------------
  float4 zv = *(const float4*)(z0 + (size_t)b * 4);
  float z[4] = {zv.x, zv.y, zv.z, zv.w};
  const float* Pb = P0 + (size_t)b * 16;
  float P[10];                                // lower triangle of symmetric P
  P[0] = Pb[0];
  P[1] = Pb[4];  P[2] = Pb[5];
  P[3] = Pb[8];  P[4] = Pb[9];  P[5] = Pb[10];
  P[6] = Pb[12]; P[7] = Pb[13]; P[8] = Pb[14]; P[9] = Pb[15];

  const int LI[10] = {0,1,1,2,2,2,3,3,3,3};
  const int LJ[10] = {0,0,1,0,1,2,0,1,2,3};

  issue_tile(0);
  issue_tile(1);

  for (int k = 0; k < NTILE; ++k) {
    // In-order async completion: with tiles k and k+1 in flight (8 ops),
    // waiting to <=4 outstanding guarantees all of tile k has landed.
    if (k < NTILE - 1) asm volatile("s_wait_asynccnt 0x4" ::: "memory");
    else               asm volatile("s_wait_asynccnt 0x0" ::: "memory");
    __syncthreads();                          // tile k visible to all waves
    const float* Ybuf = &ldsY[k & 1][0];
    __builtin_prefetch(u + (size_t)b * TLEN + (size_t)(k * TT), 0, 3);

    #pragma unroll 1
    for (int local = 0; local < TT; ++local) {
      const int t = k * TT + local;

      // ---------- sigma points: chol(P + JIT*I), cols scaled by 1.6 ----------
      float M00 = P[0]+JIT, M10 = P[1], M11 = P[2]+JIT, M20 = P[3], M21 = P[4];
      float M22 = P[5]+JIT, M30 = P[6], M31 = P[7], M32 = P[8], M33 = P[9]+JIT;
      float L00 = sqrtf(M00);          float il0 = 1.0f / L00;
      float L10 = M10*il0, L20 = M20*il0, L30 = M30*il0;
      float L11 = sqrtf(M11 - L10*L10);                  float il1 = 1.0f / L11;
      float L21 = (M21 - L20*L10)*il1, L31 = (M31 - L30*L10)*il1;
      float L22 = sqrtf(M22 - L20*L20 - L21*L21);        float il2 = 1.0f / L22;
      float L32 = (M32 - L30*L20 - L31*L21)*il2;
      float L33 = sqrtf(M33 - L30*L30 - L31*L31 - L32*L32);

      float A0[4] = {SQRT_C*L00, SQRT_C*L10, SQRT_C*L20, SQRT_C*L30};
      float A1[4] = {0.f,        SQRT_C*L11, SQRT_C*L21, SQRT_C*L31};
      float A2[4] = {0.f,        0.f,        SQRT_C*L22, SQRT_C*L32};
      float A3[4] = {0.f,        0.f,        0.f,        SQRT_C*L33};

      float Xs[9][4];
      #pragma unroll
      for (int i = 0; i < 4; ++i) {
        Xs[0][i] = z[i];
        Xs[1][i] = z[i] + A0[i]; Xs[2][i] = z[i] + A1[i];
        Xs[3][i] = z[i] + A2[i]; Xs[4][i] = z[i] + A3[i];
        Xs[5][i] = z[i] - A0[i]; Xs[6][i] = z[i] - A1[i];
        Xs[7][i] = z[i] - A2[i]; Xs[8][i] = z[i] - A3[i];
      }

      // ---------- time update (skipped at t==0, matching reference) ----------
      float zp[4];
      if (t > 0) {
        float uprev = u[(size_t)b * TLEN + (size_t)(t - 1)];
        #pragma unroll
        for (int s = 0; s < 9; ++s) {
          float p = Xs[s][0], v = Xs[s][1], kk = Xs[s][2], al = Xs[s][3];
          float p3 = p * p * p;
          Xs[s][0] = p + DTs * v;
          Xs[s][1] = v + DTs * (-CDAMP*v - kk*p - al*p3 + uprev);
        }
        #pragma unroll
        for (int i = 0; i < 4; ++i) {
          float acc = 0.f;
          #pragma unroll
          for (int s = 1; s < 9; ++s) acc += Xs[s][i];
          zp[i] = WM0 * Xs[0][i] + WMI * acc;
        }
      } else {
        zp[0] = z[0]; zp[1] = z[1]; zp[2] = z[2]; zp[3] = z[3];
      }

      // ---------- measurement sigma points ----------
      float Yv[9][2];
      #pragma unroll
      for (int s = 0; s < 9; ++s) {
        float p = Xs[s][0], v = Xs[s][1];
        Yv[s][0] = p + AP_ * p * p * p;
        Yv[s][1] = v + AV_ * v * v * v;
      }
      float yp0, yp1;
      {
        float a0 = 0.f, a1 = 0.f;
        #pragma unroll
        for (int s = 1; s < 9; ++s) { a0 += Yv[s][0]; a1 += Yv[s][1]; }
        yp0 = WM0 * Yv[0][0] + WMI * a0;
        yp1 = WM0 * Yv[0][1] + WMI * a1;
      }

      // ---------- augmented weighted Gram over sigma points ----------
      // (6x9 . 9x6 weighted Gram, 21 unique outputs; pure v_fma_f32 — WMMA
      //  block-diag packing would cost ~2x more wave-instructions here)
      float Gxx[10] = {}; float Gxy[4][2] = {}; float Gyy[3] = {};
      #pragma unroll
      for (int s = 0; s < 9; ++s) {
        float w = (s == 0) ? WC0 : WCI;
        float dx0 = Xs[s][0]-zp[0], dx1 = Xs[s][1]-zp[1];
        float dx2 = Xs[s][2]-zp[2], dx3 = Xs[s][3]-zp[3];
        float dy0 = Yv[s][0]-yp0,   dy1 = Yv[s][1]-yp1;
        float w0 = w*dx0, w1 = w*dx1, w2 = w*dx2, w3 = w*dx3, wy1 = w*dy1;
        Gxx[0] += w0*dx0; Gxx[1] += w1*dx0; Gxx[2] += w1*dx1;
        Gxx[3] += w2*dx0; Gxx[4] += w2*dx1; Gxx[5] += w2*dx2;
        Gxx[6] += w3*dx0; Gxx[7] += w3*dx1; Gxx[8] += w3*dx2; Gxx[9] += w3*dx3;
        Gxy[0][0] += w0*dy0; Gxy[0][1] += w0*dy1;
        Gxy[1][0] += w1*dy0; Gxy[1][1] += w1*dy1;
        Gxy[2][0] += w2*dy0; Gxy[2][1] += w2*dy1;
        Gxy[3][0] += w3*dy0; Gxy[3][1] += w3*dy1;
        Gyy[0] += w*dy0*dy0; Gyy[1] += wy1*dy0; Gyy[2] += wy1*dy1;
      }

      // ---------- P_pred ----------
      float Pp[10];
      if (t > 0) {
        Pp[0] = Gxx[0] + 1e-5f + JIT; Pp[1] = Gxx[1]; Pp[2] = Gxx[2] + 1e-4f + JIT;
        Pp[3] = Gxx[3]; Pp[4] = Gxx[4]; Pp[5] = Gxx[5] + 1e-6f + JIT;
        Pp[6] = Gxx[6]; Pp[7] = Gxx[7]; Pp[8] = Gxx[8]; Pp[9] = Gxx[9] + 1e-6f + JIT;
      } else {
        #pragma unroll
        for (int e = 0; e < 10; ++e) Pp[e] = P[e];
      }

      // ---------- innovation covariance + gain (2x2 exact solve) ----------
      float S00 = Gyy[0] + 0.0025f + JIT;   // S with one jitter (used in P_post)
      float S10 = Gyy[1];
      float S11 = Gyy[2] + 0.0025f + JIT;
      float C00 = S00 + JIT, C11 = S11 + JIT;   // chol(S + JIT) equivalent
      float rdet = 1.0f / (C00 * C11 - S10 * S10);

      float2 yt = *(const float2*)&Ybuf[tid * YROW + local * 2];
      float nu0 = yt.x - yp0, nu1 = yt.y - yp1;

      float K0[4], K1[4];
      #pragma unroll
      for (int i = 0; i < 4; ++i) {
        K0[i] = (Gxy[i][0]*C11 - Gxy[i][1]*S10) * rdet;
        K1[i] = (Gxy[i][1]*C00 - Gxy[i][0]*S10) * rdet;
      }

      #pragma unroll
      for (int i = 0; i < 4; ++i) z[i] = zp[i] + K0[i]*nu0 + K1[i]*nu1;

      // ---------- posterior covariance ----------
      #pragma unroll
      for (int e = 0; e < 10; ++e) {
        const int i = LI[e], j = LJ[e];
        float v = Pp[e]
                - (K0[i]*Gxy[j][0] + K1[i]*Gxy[j][1])
                - (Gxy[i][0]*K0[j] + Gxy[i][1]*K1[j])
                + S00*(K0[i]*K0[j]) + S10*(K0[i]*K1[j] + K1[i]*K0[j])
                + S11*(K1[i]*K1[j]);
        P[e] = (i == j) ? v + JIT : v;
      }

      // ---------- emit z_hist[b, t, :] (streamed once -> non-temporal) ----------
      v4f o = {z[0], z[1], z[2], z[3]};
      __builtin_nontemporal_store(o, (v4f*)(out + ((size_t)b * TLEN + (size_t)t) * 4));
    }

    __syncthreads();                 // everyone done reading buffer k&1
    if (k + 2 < NTILE) issue_tile(k + 2);   // safe to overwrite it now
  }
}

extern "C" void kernel_launch(void* const* d_in, const int* in_sizes, int n_in,
                              void* d_out, int out_size, void* d_ws, size_t ws_size,
                              hipStream_t stream) {
  const float* y  = (const float*)d_in[0];
  const float* u  = (const float*)d_in[1];
  const float* z0 = (const float*)d_in[2];
  const float* P0 = (const float*)d_in[3];
  float* out = (float*)d_out;
  const int B = in_sizes[2] / 4;            // z0 is (B, 4)
  dim3 grid(B / BLK), block(BLK);
  hipLaunchKernelGGL(ukf_kernel, grid, block, 0, stream, y, u, z0, P0, out);
}